// LatticeQuantizer_18511309046327
// MI455X (gfx1250) — compile-verified
//
#include <hip/hip_runtime.h>

// ---------------------------------------------------------------------------
// Hierarchical nested E8 lattice quantizer (encode+decode), MI455X / gfx1250.
//
// VALU-bound kernel (~2k ops/row vs 11us HBM floor at 23.3 TB/s), so the 8x8
// matmuls ride the matrix pipe: block-diagonal A2 = diag(M,M) (16x16, K=16,
// 4 chained V_WMMA_F32_16X16X4_F32) computes 32 rows per wave with the WMMA
// D-layout landing results directly in thread-per-row form for all 32 lanes.
// Cross-lane traffic is only 8 constant-pattern lane^16 swaps per matmul.
// ---------------------------------------------------------------------------

#define TINYF 1.1920928955078125e-07f   // np.finfo(np.float32).eps

typedef float v2f __attribute__((ext_vector_type(2)));
typedef float v8f __attribute__((ext_vector_type(8)));

__device__ __forceinline__ float cround(float v) {
    // floor(v - sign(v)*TINY + 0.5): round-half-toward-zero flavor.
    return floorf(v - copysignf(TINYF, v) + 0.5f);
}

// Analyze one D8 rounding branch: given x and f = cround(x), produce
//   dR   = sum (x-f)^2                       (keep plain rounding)
//   dF   = sum (x-g)^2 with worst coord flipped (g_x fix)
//   bk   = index of worst-rounded coordinate (first argmax)
//   step = +-1 flip applied at coordinate bk
// The corrected vector itself is never materialized; callers form
// y[i] = f[i] + ((useFix && i==bk) ? step : 0) on the fly.
__device__ __forceinline__ void ganal(const float* x, const float* f,
                                      float& dR, float& dF,
                                      int& bk, float& step) {
    float bd = -1.0f, bx = 0.0f, bdiff = 0.0f, ds = 0.0f;
    int k = 0;
#pragma unroll
    for (int i = 0; i < 8; ++i) {
        float diff = x[i] - f[i];
        float d = fabsf(diff);
        ds = fmaf(diff, diff, ds);
        bool gt = d > bd;                 // strict '>' keeps FIRST max
        bd    = gt ? d : bd;
        bx    = gt ? x[i] : bx;
        bdiff = gt ? diff : bdiff;        // x_k - f_k at the argmax
        k     = gt ? i : k;
    }
    // step: (xk>=0) ? (fk<xk ? +1 : -1) : (fk<=xk ? +1 : -1)
    //       with fk<xk  <=> bdiff>0,  fk<=xk <=> bdiff>=0
    step = (bx >= 0.0f) ? ((bdiff > 0.0f) ? 1.0f : -1.0f)
                        : ((bdiff >= 0.0f) ? 1.0f : -1.0f);
    float t = bdiff - step;               // residual at the flipped coord
    dR = ds;
    dF = fmaf(t, t, fmaf(-bd, bd, ds));   // ds - bd^2 + t^2
    bk = k;
}

// Closest point in E8 = D8 union (D8 + 1/2).
__device__ __forceinline__ void cpE8(const float* x, float* y) {
    float f[8], xs[8], fs[8];
    float s0 = 0.0f, s1 = 0.0f;
#pragma unroll
    for (int i = 0; i < 8; ++i) {
        f[i] = cround(x[i]);   s0 += f[i];
        xs[i] = x[i] - 0.5f;
        fs[i] = cround(xs[i]); s1 += fs[i];
    }
    float dR0, dF0, dR1, dF1, step0, step1;
    int bk0, bk1;
    ganal(x,  f,  dR0, dF0, bk0, step0);
    ganal(xs, fs, dR1, dF1, bk1, step1);
    const bool fix0 = (((int)s0) & 1) != 0;   // odd sum -> take g_x fix
    const bool fix1 = (((int)s1) & 1) != 0;
    const float d0 = fix0 ? dF0 : dR0;        // |x-y1| == |xs-fs_branch|
    const float d1 = fix1 ? dF1 : dR1;
    const bool cs = d0 < d1;
#pragma unroll
    for (int i = 0; i < 8; ++i) {
        float y0 = f[i]  + ((fix0 && i == bk0) ? step0 : 0.0f);
        float y1 = fs[i] + ((fix1 && i == bk1) ? step1 : 0.0f) + 0.5f;
        y[i] = cs ? y0 : y1;
    }
}

// Per-lane row xr[8] times an 8x8 matrix held as block-diagonal diag(M,M)
// in WMMA A-layout regs a[0..3] (K = 16, four chained K=4 WMMAs).
//
// B (4x16 f32 per slice): VGPR v, lanes 0-15: K = 4s+v,   N = lane;
//                                  lanes 16-31: K = 4s+2+v, N = lane-16.
// B columns carry rows 0-15 for K<8 and rows 16-31 for K>=8 (block diag),
// so D VGPR r gives coord r of row `lane` for ALL 32 lanes: no merge needed.
__device__ __forceinline__ void mat8_wmma(const float* xr, const v2f* a,
                                          float* yr, int half) {
    float sw[8];
#pragma unroll
    for (int i = 0; i < 8; ++i)
        sw[i] = __shfl_xor(xr[i], 16, 32);   // lane <-> lane^16 swap
    v2f b0, b1, b2, b3;
    b0.x = half ? sw[2] : xr[0];  b0.y = half ? sw[3] : xr[1];   // K 0..3
    b1.x = half ? sw[6] : xr[4];  b1.y = half ? sw[7] : xr[5];   // K 4..7
    b2.x = half ? xr[2] : sw[0];  b2.y = half ? xr[3] : sw[1];   // K 8..11
    b3.x = half ? xr[6] : sw[4];  b3.y = half ? xr[7] : sw[5];   // K 12..15
    v8f c = {};
    c = __builtin_amdgcn_wmma_f32_16x16x4_f32(false, a[0], false, b0,
                                              (short)0, c, false, false);
    c = __builtin_amdgcn_wmma_f32_16x16x4_f32(false, a[1], false, b1,
                                              (short)0, c, false, false);
    c = __builtin_amdgcn_wmma_f32_16x16x4_f32(false, a[2], false, b2,
                                              (short)0, c, false, false);
    c = __builtin_amdgcn_wmma_f32_16x16x4_f32(false, a[3], false, b3,
                                              (short)0, c, false, false);
#pragma unroll
    for (int r = 0; r < 8; ++r) yr[r] = c[r];
}

// Build block-diagonal A2 = diag(M8, M8) in WMMA A-layout.
// transposed=true  : A[m][k] = M[k][m]  (for y = x @ M  computed as M^T x^T)
// transposed=false : A[m][k] = M[m][k]  (for y = b @ M^T computed as M b^T)
template <bool TRANSPOSED>
__device__ __forceinline__ void buildA(const float* __restrict__ M,
                                       v2f* a, int m, int half) {
#pragma unroll
    for (int s = 0; s < 4; ++s) {
        const int k = 4 * s + 2 * half;          // this lane holds K=k, k+1
        const bool inblk = (s < 2) ? (m < 8) : (m >= 8);
        const int mm = m & 7;
        const int kk = k & 7;
        float vx, vy;
        if (TRANSPOSED) { vx = M[kk * 8 + mm];  vy = M[(kk + 1) * 8 + mm]; }
        else            { vx = M[mm * 8 + kk];  vy = M[mm * 8 + kk + 1];  }
        a[s].x = inblk ? vx : 0.0f;
        a[s].y = inblk ? vy : 0.0f;
    }
}

__global__ void __launch_bounds__(256)
lattice_quantizer_kernel(const float* __restrict__ x,
                         const float* __restrict__ beta_p,
                         const float* __restrict__ G,
                         const float* __restrict__ Gi,
                         const float* __restrict__ eps,
                         float* __restrict__ out, int N) {
    int row = blockIdx.x * blockDim.x + threadIdx.x;
    row = (row < N) ? row : (N - 1);          // keep EXEC all-ones for WMMA
    const int lane = threadIdx.x & 31;
    const int half = lane >> 4;
    const int m    = lane & 15;

    // One-time A-operand construction (tiny cached loads).
    v2f aGi[4], aG[4];
    buildA<true>(Gi, aGi, m, half);   // encode: y = x @ Ginv
    buildA<false>(G, aG, m, half);    // decode: y = b @ G^T

    const float beta = beta_p[0];
    const float invb = 1.0f / beta;
    float e[8];
#pragma unroll
    for (int i = 0; i < 8; ++i) e[i] = eps[i];

    // --- Load one row (two coalesced b128 loads) --------------------------
    const float4* xv = (const float4*)x;
    float4 lo = xv[2 * row + 0];
    float4 hi = xv[2 * row + 1];
    float xl[8] = {lo.x, lo.y, lo.z, lo.w, hi.x, hi.y, hi.z, hi.w};
#pragma unroll
    for (int i = 0; i < 8; ++i) xl[i] *= invb;

    // --- Encode: 3 hierarchical layers ------------------------------------
    float b[3][8];
#pragma unroll
    for (int l = 0; l < 3; ++l) {
        float t[8], y[8], v[8];
#pragma unroll
        for (int i = 0; i < 8; ++i) t[i] = xl[i] + e[i];
        cpE8(t, y);
        mat8_wmma(y, aGi, v, half);           // v = y @ Ginv (exact integers)
#pragma unroll
        for (int i = 0; i < 8; ++i) {
            // fmod(v, 4) for exact-integer floats, sign of dividend
            b[l][i] = fmaf(-4.0f, truncf(v[i] * 0.25f), v[i]);
            xl[i]   = y[i] * 0.25f;
        }
    }

    // --- Decode: accumulate residuals --------------------------------------
    float acc[8] = {0, 0, 0, 0, 0, 0, 0, 0};
    float scale = 1.0f;
#pragma unroll
    for (int l = 0; l < 3; ++l) {
        float gb[8], t[8], c[8];
        mat8_wmma(b[l], aG, gb, half);        // gb = b @ G^T
#pragma unroll
        for (int i = 0; i < 8; ++i) t[i] = gb[i] * 0.25f;
        cpE8(t, c);                            // no dither in decode
#pragma unroll
        for (int i = 0; i < 8; ++i)
            acc[i] = fmaf(scale, fmaf(-4.0f, c[i], gb[i]), acc[i]);
        scale *= 4.0f;
    }

    // --- Store (two coalesced b128 stores) ---------------------------------
    float4 o0 = {beta * acc[0], beta * acc[1], beta * acc[2], beta * acc[3]};
    float4 o1 = {beta * acc[4], beta * acc[5], beta * acc[6], beta * acc[7]};
    float4* ov = (float4*)out;
    ov[2 * row + 0] = o0;
    ov[2 * row + 1] = o1;
}

extern "C" void kernel_launch(void* const* d_in, const int* in_sizes, int n_in,
                              void* d_out, int out_size, void* d_ws, size_t ws_size,
                              hipStream_t stream) {
    const float* x    = (const float*)d_in[0];
    const float* beta = (const float*)d_in[1];
    const float* G    = (const float*)d_in[2];
    const float* Gi   = (const float*)d_in[3];
    const float* eps  = (const float*)d_in[4];
    float* out = (float*)d_out;

    const int N = in_sizes[0] / 8;            // rows
    const int block = 256;                    // 8 wave32s per workgroup
    const int grid  = (N + block - 1) / block;
    lattice_quantizer_kernel<<<grid, block, 0, stream>>>(x, beta, G, Gi, eps, out, N);
}